// RNN_87351044866828
// MI455X (gfx1250) — compile-verified
//
#include <hip/hip_runtime.h>

// Problem constants from the reference
#define TT   200      // sequence length
#define EE   512      // embed dim (== 2H, so every big GEMM has K=512)
#define HH   256      // LSTM hidden
#define G4H  1024     // 4*H
#define BB   64
#define MLPH 1024
#define NOUT 7
#define NL   2

typedef __attribute__((ext_vector_type(2))) float v2f;
typedef __attribute__((ext_vector_type(8))) float v8f;

// ---------------------------------------------------------------------------
// Embedding gather for batch row 63 only: x0[t, :] = embed[token_ids[63, t], :]
// ---------------------------------------------------------------------------
__global__ __launch_bounds__(128) void embed_row63(
    const int* __restrict__ token_ids, const float* __restrict__ embed,
    float* __restrict__ x0)
{
  int t = blockIdx.x;                 // 0..199
  int tok = token_ids[(BB - 1) * TT + t];
  const float4* src = (const float4*)(embed + (size_t)tok * EE);
  float4* dst = (float4*)(x0 + (size_t)t * EE);
  dst[threadIdx.x] = src[threadIdx.x];   // 128 * float4 = 512 floats
}

// ---------------------------------------------------------------------------
// Transpose all four whh matrices: (4, 1024, 256) -> (4, 256, 1024)
// so the scan's matvec reads coalesced columns.
// ---------------------------------------------------------------------------
__global__ __launch_bounds__(256) void transpose_whh(
    const float* __restrict__ whh, float* __restrict__ whhT)
{
  size_t idx = (size_t)blockIdx.x * 256 + threadIdx.x;   // 4*1024*256 total
  size_t ld  = idx / (G4H * HH);
  size_t rem = idx % (G4H * HH);
  size_t j = rem / HH, k = rem % HH;
  whhT[ld * (HH * G4H) + k * G4H + j] = whh[idx];
}

// ---------------------------------------------------------------------------
// Out = op(X @ W^T + bias0 + bias1), fp32 WMMA 16x16x4.
//   X: (M x K) row-major, W: (N x K) row-major, Out: (M x N)
// One wave per 32x32 output tile (2x2 grid of 16x16 WMMA accumulators).
// Per k-step (K+=4): 2 A-loads + 2 B-loads feed 4 WMMAs.
// A rows are CLAMPED (not predicated) for the M edge: garbage contributions
// land only in output rows >= M, which the epilogue never stores. This keeps
// EXEC all-ones (WMMA requirement) and avoids per-load exec save/restore.
//
// Hardware layouts (32-bit WMMA, wave32), lane = 16g + i:
//   A: {X[m][k+2g], X[m][k+2g+1]}   B tile = W^T: {W[n][k+2g], W[n][k+2g+1]}
//   D: vgpr r holds row (r + 8g), col i of the 16x16 tile.
// ---------------------------------------------------------------------------
__global__ __launch_bounds__(32) void gemm_xwt_wmma(
    const float* __restrict__ X, const float* __restrict__ W,
    const float* __restrict__ bias0, const float* __restrict__ bias1,
    float* __restrict__ Out, int M, int N, int K, int do_relu)
{
  int tilesN = N >> 5;
  int bm = blockIdx.x / tilesN;
  int bn = blockIdx.x % tilesN;
  int m0 = bm << 5, n0 = bn << 5;

  int lane = threadIdx.x;
  int g  = lane >> 4;         // half-wave select
  int i  = lane & 15;

  int rA0 = m0 + i;       if (rA0 > M - 1) rA0 = M - 1;   // clamp, don't predicate
  int rA1 = m0 + 16 + i;  if (rA1 > M - 1) rA1 = M - 1;
  const float* xrow0 = X + (size_t)rA0 * K;
  const float* xrow1 = X + (size_t)rA1 * K;
  const float* wrow0 = W + (size_t)(n0 + i) * K;
  const float* wrow1 = W + (size_t)(n0 + 16 + i) * K;

  v8f c00 = {0.f,0.f,0.f,0.f,0.f,0.f,0.f,0.f};
  v8f c01 = c00, c10 = c00, c11 = c00;

#pragma unroll 4
  for (int k = 0; k < K; k += 4) {
    int off = k + 2 * g;
    v2f a0 = *(const v2f*)(xrow0 + off);
    v2f a1 = *(const v2f*)(xrow1 + off);
    v2f b0 = *(const v2f*)(wrow0 + off);
    v2f b1 = *(const v2f*)(wrow1 + off);
    // (neg_a, A, neg_b, B, c_mod, C, reuse_a, reuse_b)
    c00 = __builtin_amdgcn_wmma_f32_16x16x4_f32(false, a0, false, b0, (short)0, c00, false, false);
    c01 = __builtin_amdgcn_wmma_f32_16x16x4_f32(false, a0, false, b1, (short)0, c01, false, false);
    c10 = __builtin_amdgcn_wmma_f32_16x16x4_f32(false, a1, false, b0, (short)0, c10, false, false);
    c11 = __builtin_amdgcn_wmma_f32_16x16x4_f32(false, a1, false, b1, (short)0, c11, false, false);
  }

  float bv0 = 0.f, bv1 = 0.f;
  if (bias0) { bv0 += bias0[n0 + i]; bv1 += bias0[n0 + 16 + i]; }
  if (bias1) { bv0 += bias1[n0 + i]; bv1 += bias1[n0 + 16 + i]; }

#pragma unroll
  for (int r = 0; r < 8; ++r) {
    int row0 = m0 + r + 8 * g;
    int row1 = row0 + 16;
    if (row0 < M) {
      float v = c00[r] + bv0; if (do_relu) v = fmaxf(v, 0.f);
      Out[(size_t)row0 * N + (n0 + i)] = v;
      v = c01[r] + bv1;       if (do_relu) v = fmaxf(v, 0.f);
      Out[(size_t)row0 * N + (n0 + 16 + i)] = v;
    }
    if (row1 < M) {
      float v = c10[r] + bv0; if (do_relu) v = fmaxf(v, 0.f);
      Out[(size_t)row1 * N + (n0 + i)] = v;
      v = c11[r] + bv1;       if (do_relu) v = fmaxf(v, 0.f);
      Out[(size_t)row1 * N + (n0 + 16 + i)] = v;
    }
  }
}

// ---------------------------------------------------------------------------
// Bidirectional LSTM scan for one layer, batch row 63 only.
// grid = 2 (blockIdx.x = direction), block = 1024 threads.
// Per step: gates[j] = xproj[t][j] + dot(whh[j,:], h)  (j = tid, 0..1023)
// using pre-transposed whhT (256 x 1024) for coalesced lane access and
// LDS-broadcast h. Threads < 256 then apply the gate nonlinearity.
// ---------------------------------------------------------------------------
__global__ __launch_bounds__(1024) void lstm_scan(
    const float* __restrict__ xproj_f,   // T x 1024 (fwd-weights projection)
    const float* __restrict__ xproj_b,   // T x 1024 (bwd-weights projection)
    const float* __restrict__ whhT_f,    // 256 x 1024
    const float* __restrict__ whhT_b,    // 256 x 1024
    float* __restrict__ y)               // T x 512 (fwd cols 0..255, bwd 256..511)
{
  int dir = blockIdx.x;
  const float* xp = dir ? xproj_b : xproj_f;
  const float* wT = dir ? whhT_b : whhT_f;

  __shared__ float h_s[HH];
  __shared__ float gates_s[G4H];

  int tid = threadIdx.x;
  if (tid < HH) h_s[tid] = 0.f;
  float ccell = 0.f;                       // valid for tid < 256
  __syncthreads();

  for (int s = 0; s < TT; ++s) {
    int t = dir ? (TT - 1 - s) : s;

    float acc = xp[(size_t)t * G4H + tid];
#pragma unroll 8
    for (int k = 0; k < HH; ++k)
      acc = fmaf(wT[(size_t)k * G4H + tid], h_s[k], acc);
    gates_s[tid] = acc;
    __syncthreads();

    if (tid < HH) {
      float ig = 1.f / (1.f + expf(-gates_s[tid]));
      float fg = 1.f / (1.f + expf(-gates_s[HH + tid]));
      float gg = tanhf(gates_s[2 * HH + tid]);
      float og = 1.f / (1.f + expf(-gates_s[3 * HH + tid]));
      ccell = fg * ccell + ig * gg;
      float hv = og * tanhf(ccell);
      h_s[tid] = hv;
      y[(size_t)t * (2 * HH) + dir * HH + tid] = hv;
    }
    __syncthreads();
  }
}

// ---------------------------------------------------------------------------
// logits = h1 @ w2^T + b2 ; out = softmax(logits) per row.
// 200 blocks x 224 threads (7 waves); wave w computes output w by warp-reduce.
// ---------------------------------------------------------------------------
__global__ __launch_bounds__(224) void logits_softmax(
    const float* __restrict__ h1, const float* __restrict__ w2,
    const float* __restrict__ b2, float* __restrict__ out)
{
  int t = blockIdx.x;
  int wv = threadIdx.x >> 5;     // 0..6
  int lane = threadIdx.x & 31;

  const float* hrow = h1 + (size_t)t * MLPH;
  const float* wrow = w2 + (size_t)wv * MLPH;
  float acc = 0.f;
  for (int k = lane; k < MLPH; k += 32)
    acc = fmaf(hrow[k], wrow[k], acc);
#pragma unroll
  for (int off = 16; off > 0; off >>= 1)
    acc += __shfl_down(acc, off, 32);

  __shared__ float logit[NOUT];
  if (lane == 0) logit[wv] = acc + b2[wv];
  __syncthreads();

  if (threadIdx.x < NOUT) {
    float mx = logit[0];
#pragma unroll
    for (int i = 1; i < NOUT; ++i) mx = fmaxf(mx, logit[i]);
    float sum = 0.f;
#pragma unroll
    for (int i = 0; i < NOUT; ++i) sum += expf(logit[i] - mx);
    out[(size_t)t * NOUT + threadIdx.x] = expf(logit[threadIdx.x] - mx) / sum;
  }
}

// ---------------------------------------------------------------------------
extern "C" void kernel_launch(void* const* d_in, const int* in_sizes, int n_in,
                              void* d_out, int out_size, void* d_ws, size_t ws_size,
                              hipStream_t stream) {
  const int*   token_ids = (const int*)  d_in[0];
  const float* embed     = (const float*)d_in[1];
  const float* wih       = (const float*)d_in[2];  // (2,2,1024,512)
  const float* whh       = (const float*)d_in[3];  // (2,2,1024,256)
  const float* bih       = (const float*)d_in[4];  // (2,2,1024)
  const float* bhh       = (const float*)d_in[5];  // (2,2,1024)
  const float* w1        = (const float*)d_in[6];  // (1024,512)
  const float* b1        = (const float*)d_in[7];  // (1024)
  const float* w2        = (const float*)d_in[8];  // (7,1024)
  const float* b2        = (const float*)d_in[9];  // (7)
  float* out = (float*)d_out;

  // Workspace layout (floats)
  float* ws = (float*)d_ws;
  float* x0     = ws;                        // 200*512
  float* xprojF = x0     + TT * EE;          // 200*1024
  float* xprojB = xprojF + TT * G4H;         // 200*1024
  float* y0     = xprojB + TT * G4H;         // 200*512
  float* y1     = y0     + TT * (2 * HH);    // 200*512
  float* whhT   = y1     + TT * (2 * HH);    // 4*256*1024
  float* h1     = whhT   + 4 * HH * G4H;     // 200*1024

  const int tilesM = (TT + 31) / 32;            // 7
  const int gemmBlocks = tilesM * (G4H / 32);   // 7*32 = 224

  // One-time-per-call prep
  transpose_whh<<<(4 * G4H * HH) / 256, 256, 0, stream>>>(whh, whhT);
  embed_row63<<<TT, 128, 0, stream>>>(token_ids, embed, x0);

  const float* layer_in = x0;
  float* layer_out[2] = { y0, y1 };

  for (int l = 0; l < NL; ++l) {
    const float* wihF = wih + (size_t)(l * 2 + 0) * G4H * EE;
    const float* wihB = wih + (size_t)(l * 2 + 1) * G4H * EE;
    const float* bihF = bih + (size_t)(l * 2 + 0) * G4H;
    const float* bihB = bih + (size_t)(l * 2 + 1) * G4H;
    const float* bhhF = bhh + (size_t)(l * 2 + 0) * G4H;
    const float* bhhB = bhh + (size_t)(l * 2 + 1) * G4H;
    const float* whhTF = whhT + (size_t)(l * 2 + 0) * HH * G4H;
    const float* whhTB = whhT + (size_t)(l * 2 + 1) * HH * G4H;

    gemm_xwt_wmma<<<gemmBlocks, 32, 0, stream>>>(
        layer_in, wihF, bihF, bhhF, xprojF, TT, G4H, EE, 0);
    gemm_xwt_wmma<<<gemmBlocks, 32, 0, stream>>>(
        layer_in, wihB, bihB, bhhB, xprojB, TT, G4H, EE, 0);

    lstm_scan<<<2, 1024, 0, stream>>>(xprojF, xprojB, whhTF, whhTB, layer_out[l]);
    layer_in = layer_out[l];
  }

  // MLP hidden: h1 = relu(y1 @ w1^T + b1)   (200x512)*(512x1024)
  gemm_xwt_wmma<<<gemmBlocks, 32, 0, stream>>>(
      y1, w1, b1, (const float*)nullptr, h1, TT, MLPH, EE, 1);

  // logits + softmax -> d_out (200 x 7)
  logits_softmax<<<TT, 224, 0, stream>>>(h1, w2, b2, out);

  (void)in_sizes; (void)n_in; (void)out_size; (void)ws_size;
}